// TwoLayerSNN_45337674776716
// MI455X (gfx1250) — compile-verified
//
#include <hip/hip_runtime.h>

// ---------------- problem constants (from reference) ----------------
#define BB 256
#define TT 1000
#define DD 32
#define HH 200
#define OO 2

// decay constants (compile-time, double->float)
#define A_SYN1  0.818730753077982f   // exp(-1/5)
#define IN_SC1  0.906346234610091f   // (1-a_syn1)*5
#define A_MEM1  0.904837418035960f   // exp(-1/10)
#define B_MEM1  0.095162581964040f   // 1-a_mem1
#define A_SYN2  0.904837418035960f   // exp(-1/10)
#define IN_SC2  0.951625819640405f   // (1-a_syn2)*10
#define A_MEM2  0.951229424500714f   // exp(-1/20)
#define B_MEM2  0.048770575499286f   // 1-a_mem2
#define TH1     0.5f
#define TH2     1.0f

typedef __attribute__((ext_vector_type(2))) float v2f;
typedef __attribute__((ext_vector_type(8))) float v8f;

// ---------------------------------------------------------------------------
// Pass 1: C1[m,h] = IN_SC1 * sum_k X[m,k] * W1[k,h]   (m = b*T + t, K = 32)
// One 16x16 output tile per wave via 8 chained V_WMMA_F32_16X16X4_F32.
// A layout (16x4 f32): lanes 0-15: M=lane, regs hold K={0,1}; lanes 16-31:
// M=lane-16, regs hold K={2,3}.  B (4x16) mirrors A with N in place of M.
// C/D layout: VGPR r -> M = r + 8*(lane>=16), N = lane&15.
// Edge tile (N columns 200..207) is handled branch-free: the load address is
// clamped in-bounds and the loaded value zeroed via select, so the inner loop
// carries no exec-mask manipulation around the WMMAs.
// ---------------------------------------------------------------------------
__global__ __launch_bounds__(128) void snn_gemm1(const float* __restrict__ X,
                                                 const float* __restrict__ W1,
                                                 float* __restrict__ C1) {
  const int  NT   = (HH + 15) / 16;                 // 13 N tiles (last is 8 wide)
  const long MT   = (long)BB * TT / 16;             // 16000 M tiles
  const int  wave = threadIdx.x >> 5;
  const int  lane = threadIdx.x & 31;
  const long tile = (long)blockIdx.x * 4 + wave;
  if (tile >= MT * NT) return;                      // uniform per wave

  const long mt   = tile / NT;
  const int  nt   = (int)(tile % NT);
  const long row0 = mt * 16;
  const int  col0 = nt * 16;

  const int  half = lane >> 4;                      // 0 -> K pair {0,1}; 1 -> {2,3}
  const long arow = row0 + (lane & 15);
  const int  bcol = col0 + (lane & 15);
  const bool colOK = bcol < HH;
  const int  bcolC = colOK ? bcol : (HH - 1);       // clamped: load always in-bounds
  const float bmask = colOK ? 1.0f : 0.0f;          // zero padded columns via select

  const float* xrow = X + arow * DD;
  const float* wcol = W1 + bcolC;

  v8f acc = {};
#pragma unroll
  for (int k0 = 0; k0 < DD; k0 += 4) {
    const int ka = k0 + half * 2;
    v2f a, b;
    a.x = xrow[ka];
    a.y = xrow[ka + 1];
    b.x = wcol[(size_t)ka * HH] * bmask;
    b.y = wcol[(size_t)(ka + 1) * HH] * bmask;
    acc = __builtin_amdgcn_wmma_f32_16x16x4_f32(
        /*neg_a=*/false, a, /*neg_b=*/false, b,
        /*c_mod=*/(short)0, acc, /*reuse_a=*/false, /*reuse_b=*/false);
  }

  const int n = col0 + (lane & 15);
  if (n < HH) {
    const long mbase = row0 + (half ? 8 : 0);
#pragma unroll
    for (int r = 0; r < 8; ++r) {
      C1[(size_t)(mbase + r) * HH + n] = acc[r] * IN_SC1;
    }
  }
}

// ---------------------------------------------------------------------------
// Pass 2: sequential scan over t. One block per batch row b; thread h owns the
// (syn1, mem1) recurrence. Layer-2 needs sum_h s1*w2[h,:] each step: wave32
// shfl_xor butterfly (5 stages) + 8-partial LDS combine; thread 0 runs the
// layer-2 recurrence for both outputs and the running average.
// c1m1 region holds C1 on entry and is overwritten with m1 (same thread, same
// element -> no hazard).
// ---------------------------------------------------------------------------
__global__ __launch_bounds__(256) void snn_scan(const float* __restrict__ W2,
                                                float* __restrict__ out0,
                                                float* __restrict__ s1o,
                                                float* c1m1,   // C1 in, m1 out (aliased)
                                                float* __restrict__ s2o,
                                                float* __restrict__ m2o) {
  const int  b      = blockIdx.x;
  const int  h      = threadIdx.x;
  const bool active = h < HH;
  const int  lane   = threadIdx.x & 31;
  const int  wid    = threadIdx.x >> 5;

  __shared__ float red[16];                         // 8 waves x 2 outputs

  float w20 = 0.0f, w21 = 0.0f;
  if (active) { w20 = W2[h * OO + 0]; w21 = W2[h * OO + 1]; }

  float syn1 = 0.0f, mem1 = 0.0f;
  float syn2a = 0.0f, syn2b = 0.0f, mem2a = 0.0f, mem2b = 0.0f;
  float acc0 = 0.0f, acc1 = 0.0f;

  size_t idx = (size_t)b * TT * HH + h;
  for (int t = 0; t < TT; ++t, idx += HH) {
    float c = 0.0f;
    if (active) {
      c = c1m1[idx];
      if (t + 1 < TT) __builtin_prefetch(&c1m1[idx + HH], 0, 0);  // global_prefetch_b8
    }
    // layer 1 LIF
    syn1 = A_SYN1 * syn1 + c;                        // c already scaled by IN_SC1
    mem1 = A_MEM1 * mem1 + B_MEM1 * syn1;
    const float s = (mem1 > TH1) ? 1.0f : 0.0f;      // heaviside(mem1 - th1)
    mem1 -= s * TH1;                                 // soft reset
    if (active) { s1o[idx] = s; c1m1[idx] = mem1; }

    // s1 @ w2 reduction across h (wave32 butterfly)
    float p0 = s * w20, p1 = s * w21;
#pragma unroll
    for (int off = 16; off > 0; off >>= 1) {
      p0 += __shfl_xor(p0, off, 32);
      p1 += __shfl_xor(p1, off, 32);
    }
    if (lane == 0) { red[wid * 2] = p0; red[wid * 2 + 1] = p1; }
    __syncthreads();

    if (threadIdx.x == 0) {
      float z0 = 0.0f, z1 = 0.0f;
#pragma unroll
      for (int w = 0; w < 8; ++w) { z0 += red[w * 2]; z1 += red[w * 2 + 1]; }
      // layer 2 LIF (O = 2, both states in thread 0)
      syn2a = A_SYN2 * syn2a + IN_SC2 * z0;
      syn2b = A_SYN2 * syn2b + IN_SC2 * z1;
      mem2a = A_MEM2 * mem2a + B_MEM2 * syn2a;
      mem2b = A_MEM2 * mem2b + B_MEM2 * syn2b;
      const float sa = (mem2a > TH2) ? 1.0f : 0.0f;
      const float sb = (mem2b > TH2) ? 1.0f : 0.0f;
      mem2a -= sa * TH2;
      mem2b -= sb * TH2;
      const size_t i2 = ((size_t)b * TT + t) * OO;
      s2o[i2] = sa;  s2o[i2 + 1] = sb;
      m2o[i2] = mem2a; m2o[i2 + 1] = mem2b;
      if (t > 0) { acc0 += mem2a; acc1 += mem2b; }   // output = mean over m2[1:]
    }
    __syncthreads();                                  // protect red[] for next t
  }

  if (threadIdx.x == 0) {
    out0[b * OO + 0] = acc0 / (float)TT;
    out0[b * OO + 1] = acc1 / (float)TT;
  }
}

extern "C" void kernel_launch(void* const* d_in, const int* in_sizes, int n_in,
                              void* d_out, int out_size, void* d_ws, size_t ws_size,
                              hipStream_t stream) {
  (void)in_sizes; (void)n_in; (void)out_size; (void)d_ws; (void)ws_size;

  const float* x  = (const float*)d_in[0];   // [B,T,D]
  const float* w1 = (const float*)d_in[1];   // [D,H]
  const float* w2 = (const float*)d_in[2];   // [H,O]

  float* out = (float*)d_out;                                  // [B,O]
  float* s1o = out + (size_t)BB * OO;                          // [B,T,H]
  float* m1o = s1o + (size_t)BB * TT * HH;                     // [B,T,H]  (C1 staged here)
  float* s2o = m1o + (size_t)BB * TT * HH;                     // [B,T,O]
  float* m2o = s2o + (size_t)BB * TT * OO;                     // [B,T,O]

  // Pass 1: C1 = IN_SC1 * (X @ W1), staged into the m1 output region.
  const long MT    = (long)BB * TT / 16;        // 16000
  const int  NT    = (HH + 15) / 16;            // 13
  const long tiles = MT * NT;                   // 208000 waves
  const long blocks = (tiles + 3) / 4;          // 4 waves per 128-thread block
  snn_gemm1<<<dim3((unsigned)blocks), dim3(128), 0, stream>>>(x, w1, m1o);

  // Pass 2: sequential scan (same stream -> ordered after pass 1).
  snn_scan<<<dim3(BB), dim3(256), 0, stream>>>(w2, out, s1o, m1o, s2o, m2o);
}